// triplet_online_74045236183447
// MI455X (gfx1250) — compile-verified
//
#include <hip/hip_runtime.h>
#include <hip/hip_bf16.h>

typedef __attribute__((ext_vector_type(2))) float v2f;
typedef __attribute__((ext_vector_type(8))) float v8f;

#define BSZ   384
#define DIM   1024
#define NTILE 24          // 384 / 16
#define MARGIN 0.5f
#define EPSV   1e-16f

// ---------------------------------------------------------------------------
// Kernel 1: squared norms  sq[i] = dot(E_i, E_i), deterministic tree reduce.
// ---------------------------------------------------------------------------
__global__ void tl_norms_kernel(const float* __restrict__ E, float* __restrict__ sq) {
    __shared__ float red[128];
    const int i   = blockIdx.x;
    const int tid = threadIdx.x;
    const float* row = E + (size_t)i * DIM;
    float s = 0.f;
    for (int k = tid; k < DIM; k += 128) { float v = row[k]; s += v * v; }
    red[tid] = s;
    __syncthreads();
    for (int off = 64; off > 0; off >>= 1) {
        if (tid < off) red[tid] += red[tid + off];
        __syncthreads();
    }
    if (tid == 0) sq[i] = red[0];
}

// ---------------------------------------------------------------------------
// Kernel 2: Gram tile via V_WMMA_F32_16X16X4_F32 (one wave per 16x16 tile),
// then convert accumulator directly to Euclidean distances.
//
// A (16x4 f32): lane L holds row (L&15); its float2 covers K = k + (L>>4)*2.
// B (4x16 f32): symmetric layout sourced from E^T, i.e. same load pattern
// from row tn*16+(L&15).  C/D: VGPR v -> M = v + 8*(L>>4), N = (L&15).
// ---------------------------------------------------------------------------
__global__ void tl_gram_dist_kernel(const float* __restrict__ E,
                                    const float* __restrict__ sq,
                                    float* __restrict__ dmat) {
    const int tile = blockIdx.x;
    const int tm   = tile / NTILE;
    const int tn   = tile % NTILE;
    const int lane = threadIdx.x;     // 0..31, wave32, EXEC all ones
    const int half = lane >> 4;       // 0 or 1
    const int l15  = lane & 15;

    const float* arow = E + (size_t)(tm * 16 + l15) * DIM + half * 2;
    const float* brow = E + (size_t)(tn * 16 + l15) * DIM + half * 2;

    v8f c = {};
#pragma unroll 4
    for (int kk = 0; kk < DIM; kk += 4) {
        v2f a = *(const v2f*)(arow + kk);   // global_load_b64, contiguous
        v2f b = *(const v2f*)(brow + kk);
        // (neg_a, A, neg_b, B, c_mod, C, reuse_a, reuse_b)
        c = __builtin_amdgcn_wmma_f32_16x16x4_f32(
                false, a, false, b, (short)0, c, false, false);
    }

    const int   n   = tn * 16 + l15;
    const float sqn = sq[n];
#pragma unroll
    for (int v = 0; v < 8; ++v) {
        const int m  = tm * 16 + v + 8 * half;
        float dd = sq[m] - 2.f * c[v] + sqn;
        dd = fmaxf(dd, 0.f);
        // reference: zm=(dd==0); d = sqrt(dd + zm*EPS) * (1-zm)
        const float dist = (dd == 0.f) ? 0.f : sqrtf(dd);
        dmat[(size_t)m * BSZ + n] = dist;
    }
}

// ---------------------------------------------------------------------------
// Kernel 3: one block per anchor a.  Only row a of the distance matrix is
// needed:  tl(a,p,n) = d[a,p] - d[a,n] + margin.
// mask = (a!=p) & (lab[a]==lab[p]) & (lab[n]!=lab[a])  (implies index
// distinctness).  Deterministic fixed-order reductions; no atomics.
// partials[a*3+{0,1,2}] = {sum_tl, num_positive, num_valid} for anchor a.
// ---------------------------------------------------------------------------
__global__ void tl_triplet_kernel(const int*   __restrict__ labels,
                                  const float* __restrict__ dmat,
                                  float*       __restrict__ partials) {
    __shared__ int   slab[BSZ];
    __shared__ float srow[BSZ];
    __shared__ float red[3][128];

    const int a   = blockIdx.x;
    const int tid = threadIdx.x;

    for (int i = tid; i < BSZ; i += 128) {
        slab[i] = labels[i];
        srow[i] = dmat[(size_t)a * BSZ + i];
    }
    __syncthreads();

    const int la = slab[a];
    float s = 0.f, np = 0.f, nv = 0.f;

    for (int p = 0; p < BSZ; ++p) {
        if (p == a || slab[p] != la) continue;     // invalid (a,p)
        const float dap = srow[p];
        for (int n = tid; n < BSZ; n += 128) {
            if (slab[n] != la) {                   // valid negative
                nv += 1.f;
                float tl = dap - srow[n] + MARGIN;
                tl = fmaxf(tl, 0.f);
                s  += tl;
                np += (tl > EPSV) ? 1.f : 0.f;
            }
        }
    }

    red[0][tid] = s; red[1][tid] = np; red[2][tid] = nv;
    __syncthreads();
    for (int off = 64; off > 0; off >>= 1) {
        if (tid < off) {
            red[0][tid] += red[0][tid + off];
            red[1][tid] += red[1][tid + off];
            red[2][tid] += red[2][tid + off];
        }
        __syncthreads();
    }
    if (tid == 0) {
        partials[a * 3 + 0] = red[0][0];
        partials[a * 3 + 1] = red[1][0];
        partials[a * 3 + 2] = red[2][0];
    }
}

// ---------------------------------------------------------------------------
// Kernel 4: fixed-order final reduction over 384 anchors -> 3 outputs.
// ---------------------------------------------------------------------------
__global__ void tl_finalize_kernel(const float* __restrict__ partials,
                                   float* __restrict__ out) {
    if (threadIdx.x == 0 && blockIdx.x == 0) {
        float S = 0.f, NP = 0.f, NV = 0.f;
        for (int a = 0; a < BSZ; ++a) {
            S  += partials[3 * a + 0];
            NP += partials[3 * a + 1];
            NV += partials[3 * a + 2];
        }
        out[0] = S / (NP + EPSV);   // loss
        out[1] = NP;                // num_positive
        out[2] = NV;                // num_valid
    }
}

// ---------------------------------------------------------------------------
extern "C" void kernel_launch(void* const* d_in, const int* in_sizes, int n_in,
                              void* d_out, int out_size, void* d_ws, size_t ws_size,
                              hipStream_t stream) {
    (void)in_sizes; (void)n_in; (void)out_size; (void)ws_size;

    const int*   labels = (const int*)d_in[0];     // [384] int32
    const float* E      = (const float*)d_in[1];   // [384,1024] f32
    float*       out    = (float*)d_out;           // 3 f32 scalars

    char*  ws       = (char*)d_ws;
    float* sq       = (float*)(ws);                          // 384 f32
    float* dmat     = (float*)(ws + 1536);                   // 384*384 f32
    float* partials = (float*)(ws + 1536 + BSZ * BSZ * 4);   // 384*3 f32

    tl_norms_kernel    <<<BSZ,           128, 0, stream>>>(E, sq);
    tl_gram_dist_kernel<<<NTILE * NTILE,  32, 0, stream>>>(E, sq, dmat);
    tl_triplet_kernel  <<<BSZ,           128, 0, stream>>>(labels, dmat, partials);
    tl_finalize_kernel <<<1,              32, 0, stream>>>(partials, out);
}